// DistributedInverseRealSHT_17549236371792
// MI455X (gfx1250) — compile-verified
//
#include <hip/hip_runtime.h>
#include <stdint.h>

// ---------------------------------------------------------------------------
// DistributedInverseRealSHT for MI455X (gfx1250, wave32, WMMA)
//
// Stage 1 (Legendre synthesis): per-m GEMM  Y[bc,k] = sum_l P[m,k,l]*X[bc,l,m]
//   -> HBM-bound on the 538MB pct table (~23us @ 23.3TB/s); fp32 WMMA 16x16x4.
//   pct tiles staged to LDS with the Tensor Data Mover (async, TENSORcnt).
// Stage 2 (irfft as DFT GEMM): out = A(8192x1088) * Basis(1088x1024),
//   A cols (2m,2m+1) = (w_m*Re, -w_m*Im); basis rows = (cos, sin)(2*pi*m*j/N).
//   Both operands L2-resident (192MB L2); fp32 WMMA again.
// ---------------------------------------------------------------------------

#define LMAXV 512
#define MMAXV 513
#define NLATV 512
#define NLONV 1024
#define CCH   16          // B*C
#define KPAD  1088        // stage-2 K: 2*513=1026 padded to 17*64
#define LCH   64          // stage-1 l-chunk

typedef float v2f __attribute__((ext_vector_type(2)));
typedef float v8f __attribute__((ext_vector_type(8)));
typedef unsigned int u32x4 __attribute__((ext_vector_type(4)));
typedef int  i32x4 __attribute__((ext_vector_type(4)));
typedef int  i32x8 __attribute__((ext_vector_type(8)));

// ---- fp32 WMMA 16x16x4: D = A(16x4) * B(4x16) + C --------------------------
__device__ __forceinline__ v8f wmma_f32(v2f a, v2f b, v8f c) {
  // 8 args: (neg_a, A, neg_b, B, c_mod, C, reuse_a, reuse_b)
  return __builtin_amdgcn_wmma_f32_16x16x4_f32(false, a, false, b,
                                               (short)0, c, false, false);
}

// ---- Tensor Data Mover: 2D tile global->LDS (D# per CDNA5 ISA) -------
// Loads tile_d1 rows of tile_d0 fp32 elements, global row stride
// row_stride elements, into LDS at lds_addr with +4 DWORD padding after every
// 64 DWORDs (=> 68-float LDS row pitch for tile_d0 == 64).
__device__ __forceinline__ void tdm_load_2d_f32(unsigned lds_addr,
                                                const void* gptr,
                                                unsigned tile_d0,
                                                unsigned tile_d1,
                                                unsigned long long row_stride) {
  uint64_t ga = (uint64_t)(uintptr_t)gptr;
  u32x4 g0;
  g0[0] = 1u;                                   // count=1, user mode
  g0[1] = lds_addr;                             // lds_addr (bytes)
  g0[2] = (unsigned)(ga & 0xffffffffu);         // global_addr[31:0]
  g0[3] = (unsigned)((ga >> 32) & 0x01ffffffu)  // global_addr[56:32]
        | (2u << 30);                           // type = 2 ("image")
  i32x8 g1;
  // data_size=2 (4B) | pad_enable | pad_interval=5 (64 DW) | pad_amount=3 (4 DW)
  g1[0] = (int)((2u << 16) | (1u << 20) | (5u << 22) | (3u << 25));
  g1[1] = (int)((tile_d0 & 0xffffu) << 16);     // tensor_dim0[15:0]
  g1[2] = (int)(((tile_d0 >> 16) & 0xffffu) | ((tile_d1 & 0xffffu) << 16));
  g1[3] = (int)(((tile_d1 >> 16) & 0xffffu) | ((tile_d0 & 0xffffu) << 16)); // tile_dim0
  g1[4] = (int)(tile_d1 & 0xffffu);             // tile_dim1 ; tile_dim2=0
  g1[5] = (int)(unsigned)(row_stride & 0xffffffffu);     // dim0_stride[31:0]
  g1[6] = (int)(unsigned)((row_stride >> 32) & 0xffffu); // dim0_stride[47:32]
  g1[7] = 0;                                    // dim1_stride unused (2D)
  i32x4 g2 = {0, 0, 0, 0};
  i32x4 g3 = {0, 0, 0, 0};
#if __clang_major__ >= 23
  i32x8 g4 = {0, 0, 0, 0, 0, 0, 0, 0};
  __builtin_amdgcn_tensor_load_to_lds(g0, g1, g2, g3, g4, 0);
#else
  __builtin_amdgcn_tensor_load_to_lds(g0, g1, g2, g3, 0);
#endif
}

__device__ __forceinline__ unsigned lds_off(const void* p) {
  // generic LDS pointer: low 32 bits are the LDS byte offset
  return (unsigned)(uintptr_t)p;
}

// ---------------------------------------------------------------------------
// Kernel 1: Legendre synthesis.  grid = (MMAX, NLAT/64), block = 128 (4 waves)
// Writes stage-2 A matrix: A[(bc*512+k)*KPAD + 2m]   =  w_m * Re
//                          A[(bc*512+k)*KPAD + 2m+1] = -w_m * Im
// ---------------------------------------------------------------------------
__global__ __launch_bounds__(128) void sht_synthesis(
    const float* __restrict__ xre, const float* __restrict__ xim,
    const float* __restrict__ pct, float* __restrict__ Aws) {
  const int m  = blockIdx.x;            // 0..512
  const int k0 = blockIdx.y * 64;       // 0..448
  const int tid  = threadIdx.x;
  const int lane = tid & 31;
  const int wave = tid >> 5;

  __shared__ float Ps[64][68];          // P tile [k][l], TDM-padded pitch 68
  __shared__ float Xs[2][LCH][17];      // X tile [re/im][l][bc]

  v8f cre = {};
  v8f cim = {};
  const float* pbase = pct + ((size_t)m * NLATV + k0) * LMAXV;

  for (int l0 = 0; l0 < LMAXV; l0 += LCH) {
    if (wave == 0) {
      tdm_load_2d_f32(lds_off(&Ps[0][0]), pbase + l0, 64u, 64u,
                      (unsigned long long)LMAXV);
    }
    // stage X[bc, l0+l, m] (strided gather; X is L2-resident at 34MB)
    for (int idx = tid; idx < 2 * LCH * CCH; idx += 128) {
      int bc = idx & 15;
      int l  = (idx >> 4) & (LCH - 1);
      int p  = idx >> 10;
      const float* src = p ? xim : xre;
      Xs[p][l][bc] = src[((size_t)bc * LMAXV + (l0 + l)) * MMAXV + m];
    }
    if (wave == 0) __builtin_amdgcn_s_wait_tensorcnt(0);
    __syncthreads();

    const int mrow = lane & 15;
    const int kh   = (lane >> 4) << 1;  // 0 or 2 (K split across lane halves)
#pragma unroll
    for (int ks = 0; ks < LCH / 4; ++ks) {
      const int l = ks * 4;
      v2f a, br, bi;
      a.x  = Ps[wave * 16 + mrow][l + kh];
      a.y  = Ps[wave * 16 + mrow][l + kh + 1];
      br.x = Xs[0][l + kh][mrow];
      br.y = Xs[0][l + kh + 1][mrow];
      bi.x = Xs[1][l + kh][mrow];
      bi.y = Xs[1][l + kh + 1][mrow];
      cre = wmma_f32(a, br, cre);
      cim = wmma_f32(a, bi, cim);
    }
    __syncthreads();
  }

  // epilogue: fold irfft weights and scatter into stage-2 A matrix
  const float wm = (m == 0 || m == MMAXV - 1) ? 1.0f : 2.0f;
  const int bc = lane & 15;             // N
  const int mh = (lane >> 4) << 3;      // M half offset
#pragma unroll
  for (int i = 0; i < 8; ++i) {
    const int k = k0 + wave * 16 + mh + i;
    float* dst = Aws + ((size_t)bc * NLATV + k) * KPAD + 2 * m;
    dst[0] = wm * cre[i];
    dst[1] = -wm * cim[i];
  }

  // m==0 blocks zero the K padding columns (2*513 .. 1087) for their rows
  if (m == 0) {
    const int npad = KPAD - 2 * MMAXV;  // 62
    for (int idx = tid; idx < CCH * 64 * npad; idx += 128) {
      int c = idx % npad;
      int r = idx / npad;
      int k = k0 + (r & 63);
      int b = r >> 6;
      Aws[((size_t)b * NLATV + k) * KPAD + 2 * MMAXV + c] = 0.0f;
    }
  }
}

// ---------------------------------------------------------------------------
// Kernel 2: DFT basis.  Basis[2m][j]=cos(2*pi*m*j/N), [2m+1][j]=sin(...),
// padding rows (>=1026) zeroed.  grid = (NLON/256, KPAD), block = 256.
// ---------------------------------------------------------------------------
__global__ void dft_basis(float* __restrict__ Bm) {
  const int j = blockIdx.x * 256 + threadIdx.x;
  const int r = blockIdx.y;
  const int mm = r >> 1;
  float v = 0.0f;
  if (mm < MMAXV) {
    const int ph = (mm * j) & (NLONV - 1);     // exact mod-N phase
    const float theta = 6.28318530717958647692f * (float)ph / (float)NLONV;
    float s, c;
    __sincosf(theta, &s, &c);
    v = (r & 1) ? s : c;
  }
  Bm[(size_t)r * NLONV + j] = v;
}

// ---------------------------------------------------------------------------
// Kernel 3: out(8192x1024) = A(8192xKPAD) * Basis(KPADx1024).
// grid = (8192/16, 1024/64), block = 128 (4 waves, one 16x16 N-subtile each).
// Both operand tiles staged via TDM.
// ---------------------------------------------------------------------------
__global__ __launch_bounds__(128) void dft_gemm(
    const float* __restrict__ Aws, const float* __restrict__ Bm,
    float* __restrict__ out) {
  const int row0 = blockIdx.x * 16;
  const int col0 = blockIdx.y * 64;
  const int tid  = threadIdx.x;
  const int lane = tid & 31;
  const int wave = tid >> 5;

  __shared__ float As[16][68];
  __shared__ float Bs[64][68];

  v8f acc = {};
  for (int kk = 0; kk < KPAD; kk += 64) {
    if (wave == 0) {
      tdm_load_2d_f32(lds_off(&As[0][0]), Aws + (size_t)row0 * KPAD + kk,
                      64u, 16u, (unsigned long long)KPAD);
      tdm_load_2d_f32(lds_off(&Bs[0][0]), Bm + (size_t)kk * NLONV + col0,
                      64u, 64u, (unsigned long long)NLONV);
      __builtin_amdgcn_s_wait_tensorcnt(0);
    }
    __syncthreads();

    const int mrow = lane & 15;
    const int kh   = (lane >> 4) << 1;
#pragma unroll
    for (int ks = 0; ks < 16; ++ks) {
      const int l = ks * 4;
      v2f a, b;
      a.x = As[mrow][l + kh];
      a.y = As[mrow][l + kh + 1];
      b.x = Bs[l + kh][wave * 16 + mrow];
      b.y = Bs[l + kh + 1][wave * 16 + mrow];
      acc = wmma_f32(a, b, acc);
    }
    __syncthreads();
  }

  const int col = col0 + wave * 16 + (lane & 15);
  const int mh  = (lane >> 4) << 3;
#pragma unroll
  for (int i = 0; i < 8; ++i) {
    const int row = row0 + mh + i;
    out[(size_t)row * NLONV + col] = acc[i];
  }
}

// ---------------------------------------------------------------------------
extern "C" void kernel_launch(void* const* d_in, const int* in_sizes, int n_in,
                              void* d_out, int out_size, void* d_ws,
                              size_t ws_size, hipStream_t stream) {
  (void)in_sizes; (void)n_in; (void)out_size; (void)ws_size;
  const float* xre = (const float*)d_in[0];
  const float* xim = (const float*)d_in[1];
  const float* pct = (const float*)d_in[2];
  // d_in[3] = nlon scalar (fixed 1024 here)
  float* out = (float*)d_out;

  float* Aws = (float*)d_ws;                               // 8192 x 1088 f32
  float* Bm  = Aws + (size_t)CCH * NLATV * KPAD;           // 1088 x 1024 f32

  sht_synthesis<<<dim3(MMAXV, NLATV / 64), 128, 0, stream>>>(xre, xim, pct, Aws);
  dft_basis<<<dim3(NLONV / 256, KPAD), 256, 0, stream>>>(Bm);
  dft_gemm<<<dim3((CCH * NLATV) / 16, NLONV / 64), 128, 0, stream>>>(Aws, Bm, out);
}